// GATlayer_34351148433852
// MI455X (gfx1250) — compile-verified
//
#include <hip/hip_runtime.h>
#include <hip/hip_bf16.h>

typedef __attribute__((ext_vector_type(2))) float v2f;
typedef __attribute__((ext_vector_type(4))) float v4f;
typedef __attribute__((ext_vector_type(8))) float v8f;
typedef __attribute__((ext_vector_type(4))) int   v4i;

#define NEG_SLOPE 0.2f
#define BN_EPS 1e-5f
typedef long long ll;

#if __has_builtin(__builtin_amdgcn_global_load_async_to_lds_b128)
#define HAVE_ASYNC_LDS 1
#else
#define HAVE_ASYNC_LDS 0
#endif

__device__ __forceinline__ void wait_async_lds() {
#if HAVE_ASYNC_LDS
#if __has_builtin(__builtin_amdgcn_s_wait_asynccnt)
  __builtin_amdgcn_s_wait_asynccnt(0);
#else
  asm volatile("s_wait_asynccnt 0x0" ::: "memory");
#endif
#endif
}

// ---------------- utility ----------------
__global__ __launch_bounds__(256) void k_fill(float* __restrict__ p, float v, ll n) {
  ll i = (ll)blockIdx.x * 256 + threadIdx.x;
  if (i < n) p[i] = v;
}

// ordered-float encoding so unsigned atomicMax == float max
__device__ __forceinline__ unsigned enc_f(float f) {
  unsigned u = __float_as_uint(f);
  return (u >> 31) ? ~u : (u | 0x80000000u);
}
__device__ __forceinline__ float dec_f(unsigned m) {
  return (m >> 31) ? __uint_as_float(m & 0x7fffffffu) : __uint_as_float(~m);
}

// ---------------- WMMA GEMM:  Y[r,c] = sum_k X[r,k]*W[c,k] (+bias[c]) ----------------
// fp32 matrix pipe (V_WMMA_F32_16X16X4_F32), wave tile 16 rows x 64 cols
// (4 accumulators reuse one A fragment). Block = 8 waves = 128 rows x 64 cols.
// W slab [64 cols x KC] double-buffered in LDS (async-to-LDS when available),
// padded stride 68 -> conflict-free b64 LDS reads.
#define KC 64
#define WS_STRIDE 68

__global__ __launch_bounds__(256) void k_gemm_xwt(
    const float* __restrict__ X, const float* __restrict__ W,
    const float* __restrict__ bias, float* __restrict__ Y,
    int nrows, int K, int ncols) {
  __shared__ float ws[2][64 * WS_STRIDE];
  const int lane = threadIdx.x & 31;
  const int wave = threadIdx.x >> 5;
  const int row0 = blockIdx.x * 128 + wave * 16;
  const int col0 = blockIdx.y * 64;
  const int mr = lane & 15;          // A-row / B-col within tile
  const int kg = (lane >> 4) << 1;   // lanes 0-15 -> k{0,1}; 16-31 -> k{2,3}
  const bool valid = (row0 < nrows);
  const float* xp = X + (ll)(valid ? row0 + mr : mr) * K + kg;

  // stage W[col0+c][k0+k] -> ws[buf][c*WS_STRIDE+k], 64x64 floats, b128 per thread
  auto stage = [&](int buf, int k0) {
#pragma unroll
    for (int it = 0; it < 4; ++it) {
      int idx = it * 256 + threadIdx.x;       // 0..1023
      int colc = idx >> 4;                    // 0..63
      int kk = (idx & 15) << 2;               // 0,4,...,60
      const float* gp = W + (ll)(col0 + colc) * K + k0 + kk;
      float* lp = &ws[buf][colc * WS_STRIDE + kk];
#if HAVE_ASYNC_LDS
      __builtin_amdgcn_global_load_async_to_lds_b128((v4i*)gp, (v4i*)lp, 0, 0);
#else
      *(v4f*)lp = *(const v4f*)gp;
#endif
    }
  };

  v8f a0 = {}, a1 = {}, a2 = {}, a3 = {};
  stage(0, 0);
  for (int k0 = 0; k0 < K; k0 += KC) {
    const int buf = (k0 / KC) & 1;
    wait_async_lds();
    __syncthreads();                  // ws[buf] ready for all waves
    if (k0 + KC < K) {
      stage(buf ^ 1, k0 + KC);
      __builtin_prefetch(xp + k0 + KC, 0, 0);   // global_prefetch_b8 on A stream
    }
    const float* bb = &ws[buf][mr * WS_STRIDE + kg];
#pragma unroll 8
    for (int k = 0; k < KC; k += 4) {
      v2f a = *(const v2f*)(xp + k0 + k);
      v2f b0 = *(const v2f*)(bb + k);
      v2f b1 = *(const v2f*)(bb + 16 * WS_STRIDE + k);
      v2f b2 = *(const v2f*)(bb + 32 * WS_STRIDE + k);
      v2f b3 = *(const v2f*)(bb + 48 * WS_STRIDE + k);
      a0 = __builtin_amdgcn_wmma_f32_16x16x4_f32(false, a, false, b0, (short)0, a0, false, false);
      a1 = __builtin_amdgcn_wmma_f32_16x16x4_f32(false, a, false, b1, (short)0, a1, false, false);
      a2 = __builtin_amdgcn_wmma_f32_16x16x4_f32(false, a, false, b2, (short)0, a2, false, false);
      a3 = __builtin_amdgcn_wmma_f32_16x16x4_f32(false, a, false, b3, (short)0, a3, false, false);
    }
    __syncthreads();                  // all waves done with ws[buf] before restage
  }

  if (!valid) return;
  // C/D layout: N = lane%16, M = reg + 8*(lane/16)
  const int srow = row0 + ((lane >> 4) << 3);
  v8f accs[4] = {a0, a1, a2, a3};
#pragma unroll
  for (int t = 0; t < 4; ++t) {
    const int scol = col0 + t * 16 + mr;
    const float bv = bias ? bias[scol] : 0.0f;
#pragma unroll
    for (int v = 0; v < 8; ++v)
      Y[(ll)(srow + v) * ncols + scol] = accs[t][v] + bv;
  }
}

// ---------------- self-loop attr: segment mean of edge_attr over dst ----------------
__global__ __launch_bounds__(256) void k_loopacc(const float* __restrict__ ea,
                                                 const int* __restrict__ dst,
                                                 float* __restrict__ loopa,
                                                 float* __restrict__ deg, ll n) {
  ll t = (ll)blockIdx.x * 256 + threadIdx.x;
  if (t >= n) return;
  int e = (int)(t >> 5), f = (int)(t & 31);
  int d = dst[e];
  atomicAdd(&loopa[(ll)d * 32 + f], ea[t]);
  if (f == 0) atomicAdd(&deg[d], 1.0f);
}
__global__ __launch_bounds__(256) void k_loopnorm(float* __restrict__ loopa,
                                                  const float* __restrict__ deg, ll n) {
  ll t = (ll)blockIdx.x * 256 + threadIdx.x;
  if (t >= n) return;
  loopa[t] /= fmaxf(deg[t >> 5], 1.0f);
}

// ---------------- attention scalars ----------------
__global__ __launch_bounds__(256) void k_attprep(const float* __restrict__ XS,
                                                 const float* __restrict__ AS,
                                                 const float* __restrict__ AD,
                                                 float* __restrict__ asrc,
                                                 float* __restrict__ adst, ll n) {
  ll t = (ll)blockIdx.x * 256 + threadIdx.x;
  if (t >= n) return;
  int node = (int)(t >> 3), h = (int)(t & 7);
  const float* xp = XS + (ll)node * 512 + h * 64;
  const float* as = AS + h * 64;
  const float* ad = AD + h * 64;
  float sa = 0.f, sd = 0.f;
#pragma unroll 8
  for (int c = 0; c < 64; ++c) { float v = xp[c]; sa += v * as[c]; sd += v * ad[c]; }
  asrc[t] = sa; adst[t] = sd;
}

// ve[h,j] = sum_c att_edge[h,c] * We[h*64+c, j]   (collapses the E'x512 edge GEMM)
__global__ __launch_bounds__(256) void k_ve(const float* __restrict__ We,
                                            const float* __restrict__ AE,
                                            float* __restrict__ VE) {
  int t = threadIdx.x;
  int h = t >> 5, j = t & 31;
  float s = 0.f;
#pragma unroll 8
  for (int c = 0; c < 64; ++c) s += AE[h * 64 + c] * We[(ll)(h * 64 + c) * 32 + j];
  VE[t] = s;
}

// ---------------- edge softmax pass 1: alpha + segment max ----------------
__global__ __launch_bounds__(256) void k_alpha1(
    const int* __restrict__ src, const int* __restrict__ dst,
    const float* __restrict__ ea, const float* __restrict__ loopa,
    const float* __restrict__ VE, const float* __restrict__ asrc,
    const float* __restrict__ adst, float* __restrict__ alpha,
    unsigned* __restrict__ mx, int E, int Etot) {
  int e = blockIdx.x * 256 + threadIdx.x;
  if (e >= Etot) return;
  int s, d; const float* ep;
  if (e < E) { s = src[e]; d = dst[e]; ep = ea + (ll)e * 32; }
  else       { s = d = e - E;          ep = loopa + (ll)(e - E) * 32; }
  float ae[8] = {0.f, 0.f, 0.f, 0.f, 0.f, 0.f, 0.f, 0.f};
  for (int c = 0; c < 32; ++c) {
    float v = ep[c];
#pragma unroll
    for (int h = 0; h < 8; ++h) ae[h] += v * VE[h * 32 + c];
  }
#pragma unroll
  for (int h = 0; h < 8; ++h) {
    float al = asrc[(ll)s * 8 + h] + adst[(ll)d * 8 + h] + ae[h];
    al = al > 0.f ? al : NEG_SLOPE * al;
    alpha[(ll)e * 8 + h] = al;
    atomicMax(&mx[(ll)d * 8 + h], enc_f(al));
  }
}

// pass 2: exp(alpha - max) + segment sum
__global__ __launch_bounds__(256) void k_alpha2(
    const int* __restrict__ dst, const unsigned* __restrict__ mx,
    float* __restrict__ alpha, float* __restrict__ den, int E, ll n) {
  ll t = (ll)blockIdx.x * 256 + threadIdx.x;
  if (t >= n) return;
  int e = (int)(t >> 3), h = (int)(t & 7);
  int d = (e < E) ? dst[e] : e - E;
  float ex = __expf(alpha[t] - dec_f(mx[(ll)d * 8 + h]));
  alpha[t] = ex;
  atomicAdd(&den[(ll)d * 8 + h], ex);
}

// pass 3: out[dst] += xs[src] * w
__global__ __launch_bounds__(256) void k_scatter(
    const int* __restrict__ src, const int* __restrict__ dst,
    const float* __restrict__ alpha, const float* __restrict__ den,
    const float* __restrict__ XS, float* __restrict__ OUT, int E, ll n) {
  ll t = (ll)blockIdx.x * 256 + threadIdx.x;
  if (t >= n) return;
  int e = (int)(t >> 6);
  int j = (int)(t & 63) << 3;
  int h = j >> 6;
  int s, d;
  if (e < E) { s = src[e]; d = dst[e]; } else { s = d = e - E; }
  float w = alpha[(ll)e * 8 + h] / (den[(ll)d * 8 + h] + 1e-16f);
  const float* xp = XS + (ll)s * 512 + j;
  float* op = OUT + (ll)d * 512 + j;
#pragma unroll
  for (int i = 0; i < 8; ++i) atomicAdd(op + i, xp[i] * w);
}

// ---------------- batchnorm (training stats) ----------------
__global__ __launch_bounds__(256) void k_bnstats(
    const float* __restrict__ X, const float* __restrict__ bias, int do_relu,
    const float* __restrict__ g, const float* __restrict__ b, int nrows,
    float* __restrict__ S, float* __restrict__ T) {
  __shared__ float ss[256], sq[256];
  int c = blockIdx.x;
  float bv = do_relu ? bias[c] : 0.f;
  float s = 0.f, q = 0.f;
  for (int r = threadIdx.x; r < nrows; r += 256) {
    float v = X[(ll)r * 512 + c];
    if (do_relu) { v += bv; v = v > 0.f ? v : 0.f; }
    s += v; q += v * v;
  }
  ss[threadIdx.x] = s; sq[threadIdx.x] = q;
  __syncthreads();
  for (int st = 128; st > 0; st >>= 1) {
    if (threadIdx.x < st) { ss[threadIdx.x] += ss[threadIdx.x + st]; sq[threadIdx.x] += sq[threadIdx.x + st]; }
    __syncthreads();
  }
  if (threadIdx.x == 0) {
    float mu = ss[0] / nrows;
    float var = sq[0] / nrows - mu * mu;
    float sc = g[c] * rsqrtf(var + BN_EPS);
    S[c] = sc; T[c] = b[c] - mu * sc;
  }
}

// x_next = bnA(relu(OUT+bias)) [+ bnB(RES)]
__global__ __launch_bounds__(256) void k_apply(
    const float* __restrict__ OUT, const float* __restrict__ bias,
    const float* __restrict__ SA, const float* __restrict__ TA,
    const float* __restrict__ RES, const float* __restrict__ SB,
    const float* __restrict__ TB, float* __restrict__ Xout, ll n) {
  ll i = (ll)blockIdx.x * 256 + threadIdx.x;
  if (i >= n) return;
  int c = (int)(i & 511);
  float v = OUT[i] + bias[c];
  v = v > 0.f ? v : 0.f;
  float r = SA[c] * v + TA[c];
  if (RES) r += SB[c] * RES[i] + TB[c];
  Xout[i] = r;
}

// ---------------- pooling + concat ----------------
__global__ __launch_bounds__(256) void k_outinit(float* __restrict__ out,
                                                 const float* __restrict__ gf, int n) {
  int t = blockIdx.x * 256 + threadIdx.x;
  if (t >= n) return;
  int g = t / 712, c = t % 712;
  out[t] = (c < 512) ? 0.0f : gf[(ll)g * 200 + (c - 512)];
}
__global__ __launch_bounds__(256) void k_pool(const float* __restrict__ X,
                                              const int* __restrict__ batch,
                                              float* __restrict__ out, ll n) {
  ll t = (ll)blockIdx.x * 256 + threadIdx.x;
  if (t >= n) return;
  int node = (int)(t >> 9), c = (int)(t & 511);
  atomicAdd(&out[(ll)batch[node] * 712 + c], X[t]);
}

// ---------------- host orchestration ----------------
extern "C" void kernel_launch(void* const* d_in, const int* in_sizes, int n_in,
                              void* d_out, int out_size, void* d_ws, size_t ws_size,
                              hipStream_t stream) {
  const float* x     = (const float*)d_in[0];
  const int*   ei    = (const int*)d_in[1];
  const float* ea    = (const float*)d_in[2];
  const int*   batch = (const int*)d_in[3];
  const float* gf    = (const float*)d_in[4];

  const int N = in_sizes[0] / 128;
  const int E = in_sizes[1] / 2;
  const int Etot = E + N;
  const int* srcp = ei;
  const int* dstp = ei + E;

  auto cw  = [&](int l, int j) { return (const float*)d_in[5 + 6 * l + j]; };
  auto bn  = [&](int l, int j) { return (const float*)d_in[29 + 2 * l + j]; };
  auto res = [&](int r, int j) { return (const float*)d_in[37 + 4 * r + j]; };

  // workspace layout (floats)
  float* XA    = (float*)d_ws;
  float* XB    = XA + (ll)N * 512;
  float* XS    = XB + (ll)N * 512;
  float* OUT   = XS + (ll)N * 512;
  float* RES   = OUT + (ll)N * 512;
  float* ALPHA = RES + (ll)N * 512;
  float* LOOPA = ALPHA + (ll)Etot * 8;
  float* DEG   = LOOPA + (ll)N * 32;
  float* ASRC  = DEG + N;
  float* ADST  = ASRC + (ll)N * 8;
  float* DEN   = ADST + (ll)N * 8;
  unsigned* MX = (unsigned*)(DEN + (ll)N * 8);
  float* VE    = (float*)(MX + (ll)N * 8);
  float* SA    = VE + 256;
  float* TA    = SA + 512;
  float* SB    = TA + 512;
  float* TB    = SB + 512;

  auto nb = [](ll n) { return (unsigned)((n + 255) / 256); };
#define L(kern, n, ...) kern<<<dim3(nb(n)), dim3(256), 0, stream>>>(__VA_ARGS__)

  // ---- self-loop edge attr (segment mean over dst) ----
  L(k_fill, (ll)N * 32, LOOPA, 0.f, (ll)N * 32);
  L(k_fill, N, DEG, 0.f, N);
  L(k_loopacc, (ll)E * 32, ea, dstp, LOOPA, DEG, (ll)E * 32);
  L(k_loopnorm, (ll)N * 32, LOOPA, DEG, (ll)N * 32);

  const dim3 ggrid((unsigned)((N + 127) / 128), 8);

  // ---- 4 GAT layers ----
  const float* Xin = x;
  float* ping[4] = {XA, XB, XA, XB};
  for (int l = 0; l < 4; ++l) {
    const int K = (l == 0) ? 128 : 512;
    k_gemm_xwt<<<ggrid, dim3(256), 0, stream>>>(Xin, cw(l, 0), nullptr, XS, N, K, 512);
    L(k_attprep, (ll)N * 8, XS, cw(l, 2), cw(l, 3), ASRC, ADST, (ll)N * 8);
    k_ve<<<dim3(1), dim3(256), 0, stream>>>(cw(l, 1), cw(l, 4), VE);
    L(k_fill, (ll)N * 8, (float*)MX, 0.f, (ll)N * 8);
    L(k_fill, (ll)N * 8, DEN, 0.f, (ll)N * 8);
    L(k_fill, (ll)N * 512, OUT, 0.f, (ll)N * 512);
    L(k_alpha1, Etot, srcp, dstp, ea, LOOPA, VE, ASRC, ADST, ALPHA, MX, E, Etot);
    L(k_alpha2, (ll)Etot * 8, dstp, MX, ALPHA, DEN, E, (ll)Etot * 8);
    L(k_scatter, (ll)Etot * 64, srcp, dstp, ALPHA, DEN, XS, OUT, E, (ll)Etot * 64);
    k_bnstats<<<dim3(512), dim3(256), 0, stream>>>(OUT, cw(l, 5), 1, bn(l, 0), bn(l, 1), N, SA, TA);
    if (l < 3) {
      k_gemm_xwt<<<ggrid, dim3(256), 0, stream>>>(Xin, res(l, 0), res(l, 1), RES, N, K, 512);
      k_bnstats<<<dim3(512), dim3(256), 0, stream>>>(RES, nullptr, 0, res(l, 2), res(l, 3), N, SB, TB);
      L(k_apply, (ll)N * 512, OUT, cw(l, 5), SA, TA, RES, SB, TB, ping[l], (ll)N * 512);
    } else {
      L(k_apply, (ll)N * 512, OUT, cw(l, 5), SA, TA, nullptr, nullptr, nullptr, ping[l], (ll)N * 512);
    }
    Xin = ping[l];
  }

  // ---- pooling + concat global features ----
  L(k_outinit, out_size, (float*)d_out, gf, out_size);
  L(k_pool, (ll)N * 512, ping[3], batch, (float*)d_out, (ll)N * 512);
#undef L
}